// EdgeConv_66133906424205
// MI455X (gfx1250) — compile-verified
//
#include <hip/hip_runtime.h>

// ---------------------------------------------------------------------------
// EdgeConv (DGCNN block) for MI455X / gfx1250, wave32, WMMA f32 16x16x4.
//
// Pipeline (all on `stream`):
//   1) knn_kernel      : per-point top-16 neighbor indices (stable ties)
//   2) zero_kernel     : clear BN stat accumulators in ws
//   3) pass<0>         : layer0 pre-BN h + stats (and shortcut stats)
//   4) finalize(0)     : BN0 + BN_sc scale/shift
//   5) pass<1>         : recompute L0(BN), L1 pre-BN + stats
//   6) finalize(1)
//   7) pass<2>         : recompute L0,L1(BN), L2 pre-BN + stats
//   8) finalize(2)
//   9) pass<3>         : full forward, K-mean, shortcut BN, ReLU, write out
// ---------------------------------------------------------------------------

#define EPSV 1e-5f

typedef float v2f __attribute__((ext_vector_type(2)));
typedef float v8f __attribute__((ext_vector_type(8)));

constexpr int kB = 128, kN = 512, kD = 16, kK = 16, kC = 64;
constexpr int kP = kB * kN;  // 65536 points

// ---------------------------------------------------------------------------
// KNN: one thread per point; coords for the batch staged in LDS; running
// top-16 insertion sort (strict > keeps stable tie order like jnp.argsort).
// ---------------------------------------------------------------------------
__global__ __launch_bounds__(128) void knn_kernel(const float* __restrict__ x,
                                                  int* __restrict__ idx) {
  __shared__ float cx[kN];
  __shared__ float cy[kN];
  __shared__ float topd[kK][128];
  __shared__ int   topi[kK][128];

  const int t  = threadIdx.x;            // 0..127
  const int b  = blockIdx.x >> 2;        // 4 blocks per batch
  const int n0 = (blockIdx.x & 3) * 128;
  const float* xb = x + (size_t)b * kN * kD;

  for (int j = t; j < kN; j += 128) {
    cx[j] = xb[(size_t)j * kD + 0];
    cy[j] = xb[(size_t)j * kD + 1];
  }
  __syncthreads();

  const int n = n0 + t;
  const float px = cx[n], py = cy[n];
#pragma unroll
  for (int k = 0; k < kK; ++k) topd[k][t] = 3.4e38f;

  for (int j = 0; j < kN; ++j) {
    float dx = px - cx[j];
    float dy = py - cy[j];
    float d2 = dx * dx + dy * dy;
    if (d2 < topd[kK - 1][t]) {
      int pos = kK - 1;
      while (pos > 0 && topd[pos - 1][t] > d2) {
        topd[pos][t] = topd[pos - 1][t];
        topi[pos][t] = topi[pos - 1][t];
        --pos;
      }
      topd[pos][t] = d2;
      topi[pos][t] = j;
    }
  }

  int* op = idx + ((size_t)(b * kN + n)) * kK;
#pragma unroll
  for (int k = 0; k < kK; ++k) op[k] = topi[k][t];
}

// ---------------------------------------------------------------------------
__global__ void zero_kernel(float* __restrict__ accum) {
  for (int i = threadIdx.x; i < 512; i += blockDim.x) accum[i] = 0.0f;
}

// ---------------------------------------------------------------------------
// One wave computes one point's conv layer: D(64x16) = W(64xIC) * H(ICx16).
// Output tiled as 4 WMMA 16x16 tiles; K-dim chained in chunks of 4 using
// V_WMMA_F32_16X16X4_F32.  A layout: lane m=l&15 holds K=4q+2*(l>>4)+{0,1}.
// B layout: lane n=l&15 holds K rows 4q+2*(l>>4)+{0,1}.  D: lane holds rows
// 16t + r + 8*(l>>4), col l&15 (r = 0..7 over the 8 acc VGPRs).
// ---------------------------------------------------------------------------
template <int IC>
__device__ __forceinline__ void conv_wmma(const float* __restrict__ W,
                                          const float* __restrict__ bias,
                                          const float* __restrict__ bnl,  // scale[64],shift[64] or null
                                          const float* hin, float* hout,
                                          int c, int hh) {
#pragma unroll
  for (int t = 0; t < 4; ++t) {
    v8f acc = {0.f, 0.f, 0.f, 0.f, 0.f, 0.f, 0.f, 0.f};
    const float* wrow = W + (size_t)(t * 16 + c) * IC + 2 * hh;
#pragma unroll
    for (int q = 0; q < IC / 4; ++q) {
      v2f a, bb;
      a[0] = wrow[4 * q];
      a[1] = wrow[4 * q + 1];
      const int r0 = 4 * q + 2 * hh;
      bb[0] = hin[r0 * 16 + c];
      bb[1] = hin[r0 * 16 + 16 + c];
      acc = __builtin_amdgcn_wmma_f32_16x16x4_f32(false, a, false, bb,
                                                  (short)0, acc, false, false);
    }
#pragma unroll
    for (int r = 0; r < 8; ++r) {
      const int ch = t * 16 + r + 8 * hh;
      float v = acc[r] + bias[ch];
      if (bnl) v = fmaxf(v * bnl[ch] + bnl[64 + ch], 0.0f);
      hout[ch * 16 + c] = v;
    }
  }
}

__device__ __forceinline__ float row_sum16(const float* r) {
  const float4* r4 = (const float4*)r;
  float s = 0.f;
#pragma unroll
  for (int i = 0; i < 4; ++i) {
    float4 v = r4[i];
    s += v.x + v.y + v.z + v.w;
  }
  return s;
}

__device__ __forceinline__ void accum_rows(const float* hx, int lane, float& s0,
                                           float& q0, float& s1, float& q1) {
  const float4* r0 = (const float4*)(hx + lane * 16);
  const float4* r1 = (const float4*)(hx + (lane + 32) * 16);
#pragma unroll
  for (int i = 0; i < 4; ++i) {
    float4 v = r0[i];
    s0 += v.x + v.y + v.z + v.w;
    q0 += v.x * v.x + v.y * v.y + v.z * v.z + v.w * v.w;
    float4 u = r1[i];
    s1 += u.x + u.y + u.z + u.w;
    q1 += u.x * u.x + u.y * u.y + u.z * u.z + u.w * u.w;
  }
}

// shortcut GEMV: lane computes channels lane and lane+32
__device__ __forceinline__ void sc_pair(const float* __restrict__ w_sc,
                                        const float* __restrict__ b_sc,
                                        const float* __restrict__ xn, int lane,
                                        float& o0, float& o1) {
  float a0 = b_sc[lane], a1 = b_sc[lane + 32];
  const float* wr0 = w_sc + (size_t)lane * kD;
  const float* wr1 = w_sc + (size_t)(lane + 32) * kD;
#pragma unroll
  for (int d = 0; d < kD; ++d) {
    float f = xn[d];
    a0 += wr0[d] * f;
    a1 += wr1[d] * f;
  }
  o0 = a0;
  o1 = a1;
}

// ---------------------------------------------------------------------------
// PHASE 0: stats for layer0 + shortcut.  PHASE 1: stats layer1.
// PHASE 2: stats layer2.  PHASE 3: final output.
// One wave per point (grid-stride).  4 waves / block, 32 KB LDS.
// ---------------------------------------------------------------------------
template <int PHASE>
__global__ __launch_bounds__(128) void pass_kernel(
    const float* __restrict__ x, const float* __restrict__ w_sc,
    const float* __restrict__ b_sc, const float* __restrict__ w0,
    const float* __restrict__ b0, const float* __restrict__ w1,
    const float* __restrict__ b1, const float* __restrict__ w2,
    const float* __restrict__ b2, const int* __restrict__ idx,
    const float* __restrict__ bnp, float* __restrict__ accum,
    float* __restrict__ out) {
  __shared__ float bufA[4][kC * kK];
  __shared__ float bufB[4][kC * kK];

  const int lane = threadIdx.x & 31;
  const int wv   = threadIdx.x >> 5;
  float* A  = bufA[wv];
  float* Bf = bufB[wv];
  const int c  = lane & 15;
  const int hh = lane >> 4;

  const int gw = blockIdx.x * 4 + wv;
  const int nW = gridDim.x * 4;

  const float* bn0 = (PHASE > 0) ? bnp : nullptr;
  const float* bn1 = (PHASE > 1) ? bnp + 128 : nullptr;
  const float* bn2 = (PHASE > 2) ? bnp + 256 : nullptr;

  float s0 = 0.f, q0 = 0.f, s1 = 0.f, q1 = 0.f;
  float ss0 = 0.f, sq0 = 0.f, ss1 = 0.f, sq1 = 0.f;

  for (int p = gw; p < kP; p += nW) {
    const int b = p >> 9;
    const int* nb = idx + (size_t)p * kK;
    const int jk = nb[c];  // this lane's neighbor (col of B)
    const int j0 = nb[0];  // nearest neighbor -> p vector

    const float* xp = x + ((size_t)(b * kN + j0)) * kD;
    const float* xk = x + ((size_t)(b * kN + jk)) * kD;

    float pv[kD];
#pragma unroll
    for (int d = 0; d < kD; ++d) pv[d] = xp[d];

    // build pairs (32 x 16) into A: rows 0..15 = p, rows 16..31 = x_knn - p
    if (hh == 0) {
#pragma unroll
      for (int d = 0; d < kD; ++d) A[d * 16 + c] = pv[d];
    } else {
#pragma unroll
      for (int d = 0; d < kD; ++d) A[(kD + d) * 16 + c] = xk[d] - pv[d];
    }

    conv_wmma<32>(w0, b0, bn0, A, Bf, c, hh);
    if (PHASE == 0) {
      accum_rows(Bf, lane, s0, q0, s1, q1);
      float a0, a1;
      sc_pair(w_sc, b_sc, x + (size_t)p * kD, lane, a0, a1);
      ss0 += a0; sq0 += a0 * a0;
      ss1 += a1; sq1 += a1 * a1;
      continue;
    }

    conv_wmma<64>(w1, b1, bn1, Bf, A, c, hh);
    if (PHASE == 1) {
      accum_rows(A, lane, s0, q0, s1, q1);
      continue;
    }

    conv_wmma<64>(w2, b2, bn2, A, Bf, c, hh);
    if (PHASE == 2) {
      accum_rows(Bf, lane, s0, q0, s1, q1);
      continue;
    }

    // PHASE 3: K-mean aggregation + BN'd shortcut + ReLU, write (B,N,C)
    float aggr0 = row_sum16(Bf + (size_t)lane * 16) * (1.0f / 16.0f);
    float aggr1 = row_sum16(Bf + (size_t)(lane + 32) * 16) * (1.0f / 16.0f);
    float a0, a1;
    sc_pair(w_sc, b_sc, x + (size_t)p * kD, lane, a0, a1);
    a0 = a0 * bnp[384 + lane] + bnp[448 + lane];
    a1 = a1 * bnp[384 + lane + 32] + bnp[448 + lane + 32];
    out[(size_t)p * kC + lane]      = fmaxf(aggr0 + a0, 0.0f);
    out[(size_t)p * kC + lane + 32] = fmaxf(aggr1 + a1, 0.0f);
  }

  if (PHASE <= 2) {
    const int base = PHASE * 128;
    atomicAdd(accum + base + lane, s0);
    atomicAdd(accum + base + 64 + lane, q0);
    atomicAdd(accum + base + lane + 32, s1);
    atomicAdd(accum + base + 64 + lane + 32, q1);
    if (PHASE == 0) {
      atomicAdd(accum + 384 + lane, ss0);
      atomicAdd(accum + 448 + lane, sq0);
      atomicAdd(accum + 384 + lane + 32, ss1);
      atomicAdd(accum + 448 + lane + 32, sq1);
    }
  }
}

// ---------------------------------------------------------------------------
// Turn sums/sumsq into per-channel BN scale/shift: y = h*scale + shift.
// layer 0 also finalizes the shortcut BN.
// ---------------------------------------------------------------------------
__global__ void finalize_kernel(const float* __restrict__ accum,
                                const float* __restrict__ g0,
                                const float* __restrict__ be0,
                                const float* __restrict__ g1,
                                const float* __restrict__ be1,
                                const float* __restrict__ g2,
                                const float* __restrict__ be2,
                                const float* __restrict__ gsc,
                                const float* __restrict__ besc,
                                float* __restrict__ bnp, int layer) {
  const int c = threadIdx.x;
  if (c >= 64) return;
  const float* g;
  const float* be;
  if (layer == 0) { g = g0; be = be0; }
  else if (layer == 1) { g = g1; be = be1; }
  else { g = g2; be = be2; }

  const float cnt = (float)((size_t)kP * kK);  // 1048576
  float mu  = accum[layer * 128 + c] / cnt;
  float var = accum[layer * 128 + 64 + c] / cnt - mu * mu;
  float rs  = rsqrtf(var + EPSV);
  float scl = rs * g[c];
  bnp[layer * 128 + c]      = scl;
  bnp[layer * 128 + 64 + c] = be[c] - mu * scl;

  if (layer == 0) {
    const float cs = (float)kP;  // 65536
    float mus  = accum[384 + c] / cs;
    float vars = accum[448 + c] / cs - mus * mus;
    float rss  = rsqrtf(vars + EPSV);
    float sscl = rss * gsc[c];
    bnp[384 + c] = sscl;
    bnp[448 + c] = besc[c] - mus * sscl;
  }
}

// ---------------------------------------------------------------------------
extern "C" void kernel_launch(void* const* d_in, const int* in_sizes, int n_in,
                              void* d_out, int out_size, void* d_ws,
                              size_t ws_size, hipStream_t stream) {
  const float* x     = (const float*)d_in[0];
  const float* w_sc  = (const float*)d_in[1];
  const float* b_sc  = (const float*)d_in[2];
  const float* g_sc  = (const float*)d_in[3];
  const float* be_sc = (const float*)d_in[4];
  const float* w0    = (const float*)d_in[5];
  const float* b0    = (const float*)d_in[6];
  const float* g0    = (const float*)d_in[7];
  const float* be0   = (const float*)d_in[8];
  const float* w1    = (const float*)d_in[9];
  const float* b1    = (const float*)d_in[10];
  const float* g1    = (const float*)d_in[11];
  const float* be1   = (const float*)d_in[12];
  const float* w2    = (const float*)d_in[13];
  const float* b2    = (const float*)d_in[14];
  const float* g2    = (const float*)d_in[15];
  const float* be2   = (const float*)d_in[16];

  // workspace layout: [0 .. P*K) int idx | [P*K .. +512) accum | +512 bnp[512]
  int*   idx   = (int*)d_ws;
  float* accum = (float*)d_ws + (size_t)kP * kK;
  float* bnp   = accum + 512;
  float* out   = (float*)d_out;

  knn_kernel<<<kB * 4, 128, 0, stream>>>(x, idx);
  zero_kernel<<<1, 256, 0, stream>>>(accum);

  pass_kernel<0><<<1024, 128, 0, stream>>>(x, w_sc, b_sc, w0, b0, w1, b1, w2,
                                           b2, idx, bnp, accum, out);
  finalize_kernel<<<1, 64, 0, stream>>>(accum, g0, be0, g1, be1, g2, be2, g_sc,
                                        be_sc, bnp, 0);
  pass_kernel<1><<<1024, 128, 0, stream>>>(x, w_sc, b_sc, w0, b0, w1, b1, w2,
                                           b2, idx, bnp, accum, out);
  finalize_kernel<<<1, 64, 0, stream>>>(accum, g0, be0, g1, be1, g2, be2, g_sc,
                                        be_sc, bnp, 1);
  pass_kernel<2><<<1024, 128, 0, stream>>>(x, w_sc, b_sc, w0, b0, w1, b1, w2,
                                           b2, idx, bnp, accum, out);
  finalize_kernel<<<1, 64, 0, stream>>>(accum, g0, be0, g1, be1, g2, be2, g_sc,
                                        be_sc, bnp, 2);
  pass_kernel<3><<<4096, 128, 0, stream>>>(x, w_sc, b_sc, w0, b0, w1, b1, w2,
                                           b2, idx, bnp, accum, out);
}